// Discriminator_67688684585267
// MI455X (gfx1250) — compile-verified
//
#include <hip/hip_runtime.h>
#include <math.h>

// ---- problem constants (match reference) ----
#define SEQ   512
#define BATCH 128
#define H     100
#define HP    112          // H padded to multiple of 16 (7 col tiles)
#define LDSP  113          // LDS row stride (odd -> bank-conflict free)
#define NLAY  10
#define KT    (HP / 4)     // 28 K-tiles of 4 for f32 WMMA
#define CT    (HP / 16)    // 7 column tiles

typedef __attribute__((ext_vector_type(2))) float v2f;
typedef __attribute__((ext_vector_type(8))) float v8f;

static __device__ __forceinline__
v8f wmma_f32_16x16x4(v2f a, v2f b, v8f c) {
    // D = A(16x4) * B(4x16) + C(16x16), all f32, wave32
    return __builtin_amdgcn_wmma_f32_16x16x4_f32(
        /*neg_a=*/false, a, /*neg_b=*/false, b,
        /*c_mod=*/(short)0, c, /*reuse_a=*/false, /*reuse_b=*/false);
}

// Hardware transcendental tanh (CDNA5 V_TANH_F32). Trailing v_nop covers the
// TRANS-op result hazard (ISA 7.4: one independent op / V_NOP before use).
static __device__ __forceinline__ float hw_tanh(float x) {
    float r;
    asm volatile("v_tanh_f32 %0, %1\n\tv_nop" : "=v"(r) : "v"(x));
    return r;
}

// ---------------------------------------------------------------------------
// Pad x (SEQ*BATCH, H) -> (SEQ*BATCH, HP) with zeros in cols [H, HP)
// ---------------------------------------------------------------------------
__global__ void pad_input(const float* __restrict__ x, float* __restrict__ xp) {
    int e = blockIdx.x * blockDim.x + threadIdx.x;
    const int total = SEQ * BATCH * HP;
    if (e >= total) return;
    int row = e / HP;
    int j   = e - row * HP;
    xp[e] = (j < H) ? x[row * H + j] : 0.0f;
}

// ---------------------------------------------------------------------------
// pre = seq_in @ W_ih^T + (b_ih + b_hh)   (per layer)
// grid.x = (SEQ*BATCH)/16 row tiles, block = 7 waves (one per col tile)
// ---------------------------------------------------------------------------
__global__ void pre_gemm(const float* __restrict__ xp,   // [SEQ*BATCH][HP]
                         const float* __restrict__ Wih,  // [H][H] row major
                         const float* __restrict__ bih,  // [H]
                         const float* __restrict__ bhh,  // [H]
                         float* __restrict__ pre)        // [SEQ*BATCH][HP]
{
    __shared__ float atile[16][LDSP];

    const int rt   = blockIdx.x;          // row tile (16 rows of T*B)
    const int wave = threadIdx.x >> 5;    // 0..6 = col tile
    const int lane = threadIdx.x & 31;
    const int m    = lane & 15;           // A-row / C-col index
    const int half = lane >> 4;           // selects K pair {0,1} vs {2,3}
    const int jt   = wave;
    const int j    = jt * 16 + m;         // output column (B/C "N")

    // Stage A tile (16 x HP) into LDS cooperatively
    const size_t base = (size_t)rt * 16 * HP;
    for (int e = threadIdx.x; e < 16 * HP; e += blockDim.x) {
        int r = e / HP, c = e - r * HP;
        atile[r][c] = xp[base + e];
    }
    __syncthreads();

    // B fragments of W_ih^T for this column tile (register resident)
    v2f breg[KT];
#pragma unroll
    for (int kt = 0; kt < KT; ++kt) {
        int k = kt * 4 + half * 2;
        breg[kt].x = (j < H && k     < H) ? Wih[j * H + k]     : 0.0f;
        breg[kt].y = (j < H && k + 1 < H) ? Wih[j * H + k + 1] : 0.0f;
    }

    const float bias = (j < H) ? (bih[j] + bhh[j]) : 0.0f;
    v8f c;
#pragma unroll
    for (int v = 0; v < 8; ++v) c[v] = bias;

#pragma unroll
    for (int kt = 0; kt < KT; ++kt) {
        int k = kt * 4 + half * 2;
        v2f a; a.x = atile[m][k]; a.y = atile[m][k + 1];
        c = wmma_f32_16x16x4(a, breg[kt], c);
    }

    // C layout: VGPR v -> row (v + half*8), col m
    float* ob = pre + base + (size_t)jt * 16 + m;
#pragma unroll
    for (int v = 0; v < 8; ++v)
        ob[(size_t)(v + half * 8) * HP] = c[v];
}

// ---------------------------------------------------------------------------
// Sequential scan: h_t = tanh(pre_t + h_{t-1} @ W_hh^T), write out seq.
// grid.x = BATCH/16 (independent batch slices), block = 7 waves.
// h double-buffered in LDS; W_hh^T fragments register resident; the next
// step's pre tile is prefetched into registers during the WMMA chain.
// ---------------------------------------------------------------------------
__global__ void rnn_scan(const float* __restrict__ pre,  // [SEQ][BATCH][HP]
                         const float* __restrict__ Whh,  // [H][H]
                         float* __restrict__ outp)       // [SEQ][BATCH][HP]
{
    __shared__ float hbuf[2][16][LDSP];

    const int bt   = blockIdx.x;          // batch tile
    const int wave = threadIdx.x >> 5;
    const int lane = threadIdx.x & 31;
    const int m    = lane & 15;
    const int half = lane >> 4;
    const int jt   = wave;
    const int j    = jt * 16 + m;

    // W_hh^T fragments in registers (28 x v2f = 56 VGPRs)
    v2f breg[KT];
#pragma unroll
    for (int kt = 0; kt < KT; ++kt) {
        int k = kt * 4 + half * 2;
        breg[kt].x = (j < H && k     < H) ? Whh[j * H + k]     : 0.0f;
        breg[kt].y = (j < H && k + 1 < H) ? Whh[j * H + k + 1] : 0.0f;
    }

    // h_0 = 0
    for (int e = threadIdx.x; e < 2 * 16 * LDSP; e += blockDim.x)
        (&hbuf[0][0][0])[e] = 0.0f;
    __syncthreads();

    // Prefetch pre tile for t = 0
    const size_t tileOff = (size_t)bt * 16 * HP + (size_t)jt * 16 + m;
    float cn[8];
    {
        const float* pb = pre + tileOff;
#pragma unroll
        for (int v = 0; v < 8; ++v) cn[v] = pb[(size_t)(v + half * 8) * HP];
    }

    for (int t = 0; t < SEQ; ++t) {
        const float (*ht)[LDSP] = hbuf[t & 1];

        // C init from prefetched pre tile; two independent accumulation chains
        v8f c0, c1;
#pragma unroll
        for (int v = 0; v < 8; ++v) { c0[v] = cn[v]; c1[v] = 0.0f; }

        // Prefetch next step's pre tile; its latency hides under the WMMAs
        if (t + 1 < SEQ) {
            const float* pb = pre + (size_t)(t + 1) * BATCH * HP + tileOff;
#pragma unroll
            for (int v = 0; v < 8; ++v) cn[v] = pb[(size_t)(v + half * 8) * HP];
        }

#pragma unroll
        for (int kt = 0; kt < KT; kt += 2) {
            int k0 = kt * 4 + half * 2;
            v2f a0; a0.x = ht[m][k0]; a0.y = ht[m][k0 + 1];
            c0 = wmma_f32_16x16x4(a0, breg[kt], c0);
            int k1 = (kt + 1) * 4 + half * 2;
            v2f a1; a1.x = ht[m][k1]; a1.y = ht[m][k1 + 1];
            c1 = wmma_f32_16x16x4(a1, breg[kt + 1], c1);
        }

        float (*hn)[LDSP] = hbuf[(t + 1) & 1];
        float* ob = outp + ((size_t)t * BATCH + bt * 16) * HP
                         + (size_t)jt * 16 + m;
#pragma unroll
        for (int v = 0; v < 8; ++v) {
            float val = hw_tanh(c0[v] + c1[v]);
            int row = v + half * 8;
            hn[row][jt * 16 + m] = val;
            ob[(size_t)row * HP]  = val;
        }
        __syncthreads();   // writes to hn visible before next step's reads
    }
}

// ---------------------------------------------------------------------------
// out[i] = sigmoid(dot(seq[i][0:H], W_lin) + b_lin)
// ---------------------------------------------------------------------------
__global__ void head_kernel(const float* __restrict__ seq,   // [SEQ*BATCH][HP]
                            const float* __restrict__ Wlin,  // [H]
                            const float* __restrict__ blin,  // [1]
                            float* __restrict__ out)
{
    int idx = blockIdx.x * blockDim.x + threadIdx.x;
    if (idx >= SEQ * BATCH) return;
    const float* row = seq + (size_t)idx * HP;
    float acc = blin[0];
#pragma unroll 4
    for (int i = 0; i < H; ++i) acc += row[i] * Wlin[i];
    out[idx] = 1.0f / (1.0f + expf(-acc));
}

// ---------------------------------------------------------------------------
extern "C" void kernel_launch(void* const* d_in, const int* in_sizes, int n_in,
                              void* d_out, int out_size, void* d_ws, size_t ws_size,
                              hipStream_t stream) {
    const float* x     = (const float*)d_in[0];
    const float* W_ih  = (const float*)d_in[1];   // [L][H][H]
    const float* W_hh  = (const float*)d_in[2];   // [L][H][H]
    const float* b_ih  = (const float*)d_in[3];   // [L][H]
    const float* b_hh  = (const float*)d_in[4];   // [L][H]
    const float* W_lin = (const float*)d_in[5];   // [H]
    const float* b_lin = (const float*)d_in[6];   // [1]
    float* out = (float*)d_out;

    const size_t N = (size_t)SEQ * BATCH * HP;    // padded buffer elems
    float* bufA = (float*)d_ws;                   // layer input  (padded)
    float* bufB = bufA + N;                       // layer output (padded)
    float* preb = bufB + N;                       // pre-activations
    (void)in_sizes; (void)n_in; (void)out_size; (void)ws_size;

    // pad x into bufA
    {
        int total = SEQ * BATCH * HP;
        pad_input<<<(total + 255) / 256, 256, 0, stream>>>(x, bufA);
    }

    float* cur = bufA;
    float* nxt = bufB;
    const int rowTiles = (SEQ * BATCH) / 16;      // 4096
    for (int l = 0; l < NLAY; ++l) {
        const float* Wih = W_ih + (size_t)l * H * H;
        const float* Whh = W_hh + (size_t)l * H * H;
        const float* bih = b_ih + (size_t)l * H;
        const float* bhh = b_hh + (size_t)l * H;

        pre_gemm<<<rowTiles, CT * 32, 0, stream>>>(cur, Wih, bih, bhh, preb);
        rnn_scan<<<BATCH / 16, CT * 32, 0, stream>>>(preb, Whh, nxt);

        float* tmp = cur; cur = nxt; nxt = tmp;
    }

    head_kernel<<<(SEQ * BATCH + 127) / 128, 128, 0, stream>>>(cur, W_lin, b_lin, out);
}